// LSA_56178172231776
// MI455X (gfx1250) — compile-verified
//
#include <hip/hip_runtime.h>

// Problem constants (match reference)
#define B_     16
#define N_     1024
#define DIM_   768
#define HEADS_ 12
#define DHEAD_ 64
#define INNER_ 768          // HEADS*DHEAD
#define QKV3_  (3 * INNER_) // 2304

typedef __attribute__((ext_vector_type(2))) float v2f;
typedef __attribute__((ext_vector_type(8))) float v8f;
typedef int v4i_b __attribute__((vector_size(16)));   // matches builtin param type

#define GLOBAL_AS __attribute__((address_space(1)))
#define LDS_AS    __attribute__((address_space(3)))

static __device__ __forceinline__ v8f wmma_f32(v2f a, v2f b, v8f c) {
    // D = A(16x4) * B(4x16) + C(16x16), fp32 matrix pipe on gfx1250
    return __builtin_amdgcn_wmma_f32_16x16x4_f32(
        /*neg_a=*/false, a, /*neg_b=*/false, b,
        /*c_mod=*/(short)0, c, /*reuse_a=*/false, /*reuse_b=*/false);
}

// 16-byte global -> LDS copy via the CDNA5 async-LDS path (ASYNCcnt-tracked,
// no VGPR round-trip). Builtin signature (from compiler diagnostic):
//   (int4 __device__* src_global, int4 __shared__* dst_lds, imm offset, imm cpol)
static __device__ __forceinline__ void async_copy16(const float* g, float* l) {
#if defined(__has_builtin) && __has_builtin(__builtin_amdgcn_global_load_async_to_lds_b128)
    __builtin_amdgcn_global_load_async_to_lds_b128(
        (GLOBAL_AS v4i_b*)g, (LDS_AS v4i_b*)l, /*offset=*/0, /*cpol=*/0);
#else
    *(float4*)l = *(const float4*)g;
#endif
}

static __device__ __forceinline__ void wait_async_lds() {
#if defined(__has_builtin) && __has_builtin(__builtin_amdgcn_s_wait_asynccnt)
    __builtin_amdgcn_s_wait_asynccnt(0);
#else
    asm volatile("s_wait_asynccnt 0x0" ::: "memory");
#endif
}

// ---------------------------------------------------------------------------
// Tiled fp32 WMMA GEMM: C[M,N] = A[M,K] * B[K,N] (+ bias[N] if bias != null)
// Block = 128(M) x 256(N) tile, 256 threads = 8 waves; each wave owns a
// 64x64 patch (4x4 WMMA tiles -> 16 WMMAs per k-step from 8 frag loads).
// K staged in 32-chunks, double-buffered LDS filled by async-LDS loads.
// Requires M%128==0, N%256==0, K%32==0 (true for all GEMMs here).
// ---------------------------------------------------------------------------
__global__ __launch_bounds__(256) void gemm_wmma_k(
    const float* __restrict__ A, const float* __restrict__ Bm,
    const float* __restrict__ bias, float* __restrict__ C,
    int M, int N, int K)
{
    __shared__ float As[2][128 * 36];   // 128 x 32, stride 36 (16B-aligned rows)
    __shared__ float Bs[2][32 * 260];   // 32 x 256, stride 260

    const int tid   = threadIdx.x;
    const int lane  = tid & 31;
    const int wave  = tid >> 5;
    const int col   = lane & 15;
    const int halfw = lane >> 4;

    const int m0 = blockIdx.y * 128;
    const int n0 = blockIdx.x * 256;
    const int wm = (wave >> 2) * 64;   // 0 / 64
    const int wn = (wave & 3) * 64;    // 0/64/128/192

    const v8f vzero = {0.f, 0.f, 0.f, 0.f, 0.f, 0.f, 0.f, 0.f};
    v8f acc[4][4];
#pragma unroll
    for (int i = 0; i < 4; ++i)
#pragma unroll
        for (int j = 0; j < 4; ++j) acc[i][j] = vzero;

    // async stage of one K-chunk into buffer `buf`
    auto stage = [&](int buf, int kk) {
        const int ra  = tid >> 3;          // 0..31
        const int k4  = (tid & 7) * 4;     // 0..28
#pragma unroll
        for (int p = 0; p < 4; ++p) {
            const int row = p * 32 + ra;
            async_copy16(&A[(size_t)(m0 + row) * K + kk + k4],
                         &As[buf][row * 36 + k4]);
        }
        const int rb  = tid >> 6;          // 0..3
        const int n4  = (tid & 63) * 4;    // 0..252
#pragma unroll
        for (int p = 0; p < 8; ++p) {
            const int row = p * 4 + rb;
            async_copy16(&Bm[(size_t)(kk + row) * N + n0 + n4],
                         &Bs[buf][row * 260 + n4]);
        }
    };

    const int nchunks = K / 32;
    stage(0, 0);
    wait_async_lds();
    __syncthreads();

    int buf = 0;
    for (int c = 0; c < nchunks; ++c) {
        if (c + 1 < nchunks) stage(buf ^ 1, (c + 1) * 32);  // overlap with WMMAs

#pragma unroll
        for (int s = 0; s < 8; ++s) {
            const int kb = s * 4 + 2 * halfw;
            v2f af[4];
#pragma unroll
            for (int i = 0; i < 4; ++i)
                af[i] = *(const v2f*)&As[buf][(wm + i * 16 + col) * 36 + kb];
            v2f bf[4];
#pragma unroll
            for (int j = 0; j < 4; ++j) {
                v2f b;
                b.x = Bs[buf][(kb)     * 260 + wn + j * 16 + col];
                b.y = Bs[buf][(kb + 1) * 260 + wn + j * 16 + col];
                bf[j] = b;
            }
#pragma unroll
            for (int i = 0; i < 4; ++i)
#pragma unroll
                for (int j = 0; j < 4; ++j)
                    acc[i][j] = wmma_f32(af[i], bf[j], acc[i][j]);
        }

        if (c + 1 < nchunks) wait_async_lds();
        __syncthreads();
        buf ^= 1;
    }

    // Write back (C layout: M = e + 8*half, N = col)
#pragma unroll
    for (int j = 0; j < 4; ++j) {
        const int cidx = n0 + wn + j * 16 + col;
        const float bv = bias ? bias[cidx] : 0.0f;
#pragma unroll
        for (int i = 0; i < 4; ++i) {
#pragma unroll
            for (int e = 0; e < 8; ++e) {
                const int r = m0 + wm + i * 16 + e + 8 * halfw;
                C[(size_t)r * N + cidx] = acc[i][j][e] + bv;
            }
        }
    }
}

// ---------------------------------------------------------------------------
// Flash-style attention with diagonal self-mask.
// qkv layout: [B, N, 3*INNER] = [.., q(h,d) | k(h,d) | v(h,d)]
// Block = one (b, head) x 64 query rows; 4 waves, 16 query rows each.
// K/V streamed 128 rows at a time through double-buffered LDS (async-LDS
// loads overlapped with WMMAs); online softmax; Q@K^T and P@V on
// v_wmma_f32_16x16x4_f32. Output: attn_out [B, N, INNER].
// ---------------------------------------------------------------------------
__global__ __launch_bounds__(128) void attn_wmma_kernel(
    const float* __restrict__ qkv, float* __restrict__ outp)
{
    __shared__ float Ks[2][128 * 68];   // 128 keys x 64 d, stride 68
    __shared__ float Vs[2][128 * 68];
    __shared__ float Pt[4][16 * 18];    // per-wave P tile, stride 18

    const int tid  = threadIdx.x;
    const int lane = tid & 31;
    const int wave = tid >> 5;     // 0..3
    const int col  = lane & 15;
    const int hf   = lane >> 4;

    const int b    = blockIdx.x / HEADS_;
    const int head = blockIdx.x % HEADS_;
    const int q0   = blockIdx.y * 64 + wave * 16;

    const float* qbase = qkv + (size_t)b * N_ * QKV3_ + head * DHEAD_;
    const float* kbase = qbase + INNER_;
    const float* vbase = qbase + 2 * INNER_;

    auto stage_kv = [&](int buf, int kb) {
        const int r  = tid >> 4;         // 0..7
        const int d4 = (tid & 15) * 4;   // 0..60
#pragma unroll
        for (int p = 0; p < 16; ++p) {
            const int row = p * 8 + r;
            async_copy16(&kbase[(size_t)(kb + row) * QKV3_ + d4],
                         &Ks[buf][row * 68 + d4]);
            async_copy16(&vbase[(size_t)(kb + row) * QKV3_ + d4],
                         &Vs[buf][row * 68 + d4]);
        }
    };

    // Q fragments (A layout): row = q0+col, K index = 4s + 2*hf
    v2f qa[16];
    {
        const float* qr = qbase + (size_t)(q0 + col) * QKV3_ + 2 * hf;
#pragma unroll
        for (int s = 0; s < 16; ++s)
            qa[s] = *(const v2f*)&qr[s * 4];
    }

    const v8f vzero = {0.f, 0.f, 0.f, 0.f, 0.f, 0.f, 0.f, 0.f};
    v8f acc_o[4];
#pragma unroll
    for (int t = 0; t < 4; ++t) acc_o[t] = vzero;
    float mrun[8], lrun[8];
#pragma unroll
    for (int e = 0; e < 8; ++e) { mrun[e] = -3.0e38f; lrun[e] = 0.0f; }

    stage_kv(0, 0);
    wait_async_lds();
    __syncthreads();

    int buf = 0;
    for (int c = 0; c < N_ / 128; ++c) {
        const int kb = c * 128;
        if (c + 1 < N_ / 128) stage_kv(buf ^ 1, kb + 128);  // overlap

        for (int j = 0; j < 8; ++j) {
            // S = Q * K^T  (16x16 scores over 16 keys), K-dim 64 -> 16 WMMA
            v8f s8 = vzero;
#pragma unroll
            for (int s = 0; s < 16; ++s) {
                const v2f bfr =
                    *(const v2f*)&Ks[buf][(j * 16 + col) * 68 + s * 4 + 2 * hf];
                s8 = wmma_f32(qa[s], bfr, s8);
            }

            // scale + diagonal mask + online softmax (rows live in lane halves)
            float p[8], alpha[8];
#pragma unroll
            for (int e = 0; e < 8; ++e) {
                float sv = s8[e] * 0.125f;             // dim_head^-0.5
                const int qi = q0 + e + 8 * hf;
                const int kj = kb + j * 16 + col;
                if (qi == kj) sv = -3.0e38f;           // self-mask (lane select)
                float v = sv;
#pragma unroll
                for (int off = 1; off < 16; off <<= 1)
                    v = fmaxf(v, __shfl_xor(v, off, 32));
                const float mn = fmaxf(mrun[e], v);
                alpha[e] = __expf(mrun[e] - mn);
                mrun[e] = mn;
                const float pe = __expf(sv - mn);
                p[e] = pe;
                float rs = pe;
#pragma unroll
                for (int off = 1; off < 16; off <<= 1)
                    rs += __shfl_xor(rs, off, 32);
                lrun[e] = lrun[e] * alpha[e] + rs;
            }
#pragma unroll
            for (int t = 0; t < 4; ++t)
#pragma unroll
                for (int e = 0; e < 8; ++e)
                    acc_o[t][e] *= alpha[e];

            // C-layout P -> A-layout via per-wave LDS round trip
#pragma unroll
            for (int e = 0; e < 8; ++e)
                Pt[wave][(e + 8 * hf) * 18 + col] = p[e];
            v2f pa[4];
#pragma unroll
            for (int s = 0; s < 4; ++s)
                pa[s] = *(const v2f*)&Pt[wave][col * 18 + s * 4 + 2 * hf];

            // O += P * V  (K-dim 16 -> 4 WMMA per 16-wide d-tile)
#pragma unroll
            for (int t = 0; t < 4; ++t) {
#pragma unroll
                for (int s = 0; s < 4; ++s) {
                    v2f bfr;
                    bfr.x = Vs[buf][(j * 16 + s * 4 + 2 * hf)     * 68 + t * 16 + col];
                    bfr.y = Vs[buf][(j * 16 + s * 4 + 2 * hf + 1) * 68 + t * 16 + col];
                    acc_o[t] = wmma_f32(pa[s], bfr, acc_o[t]);
                }
            }
        }

        if (c + 1 < N_ / 128) wait_async_lds();
        __syncthreads();
        buf ^= 1;
    }

    // Normalize and write attn_out [B, N, INNER]
#pragma unroll
    for (int e = 0; e < 8; ++e) {
        const float inv = 1.0f / lrun[e];
        const int orow = q0 + e + 8 * hf;
        float* op = outp + ((size_t)(b * N_ + orow)) * INNER_ + head * DHEAD_;
#pragma unroll
        for (int t = 0; t < 4; ++t)
            op[t * 16 + col] = acc_o[t][e] * inv;
    }
}

// ---------------------------------------------------------------------------
// Launch: qkv GEMM -> flash attention -> output GEMM (+bias)
// Workspace: qkv [16384 x 2304] f32 (151 MB) + attn [16384 x 768] f32 (48 MB)
// ---------------------------------------------------------------------------
extern "C" void kernel_launch(void* const* d_in, const int* in_sizes, int n_in,
                              void* d_out, int out_size, void* d_ws, size_t ws_size,
                              hipStream_t stream) {
    (void)in_sizes; (void)n_in; (void)out_size; (void)ws_size;
    const float* x     = (const float*)d_in[0];
    const float* w_qkv = (const float*)d_in[1];
    const float* w_out = (const float*)d_in[2];
    const float* b_out = (const float*)d_in[3];
    float* out  = (float*)d_out;
    float* qkv  = (float*)d_ws;
    float* attn = qkv + (size_t)(B_ * N_) * QKV3_;

    const int M = B_ * N_;  // 16384

    dim3 g1(QKV3_ / 256, M / 128);   // 9 x 128
    gemm_wmma_k<<<g1, dim3(256), 0, stream>>>(x, w_qkv, nullptr, qkv, M, QKV3_, DIM_);

    dim3 g2(B_ * HEADS_, N_ / 64);   // 192 x 16
    attn_wmma_kernel<<<g2, dim3(128), 0, stream>>>(qkv, attn);

    dim3 g3(DIM_ / 256, M / 128);    // 3 x 128
    gemm_wmma_k<<<g3, dim3(256), 0, stream>>>(attn, w_out, b_out, out, M, DIM_, INNER_);
}